// MeshDeformationModel_89730456748236
// MI455X (gfx1250) — compile-verified
//
#include <hip/hip_runtime.h>
#include <math.h>

typedef float v2f __attribute__((ext_vector_type(2)));
typedef float v8f __attribute__((ext_vector_type(8)));

#define DM 256

// ---------------------------------------------------------------------------
// WMMA GEMM: C[n,N] = A[n,K] * W^T   (W row-major [N,K]), + bias, opt relu,
// opt residual add. One wave computes a 16x64 strip = 4 x (16x16) tiles via
// V_WMMA_F32_16X16X4_F32, sharing one A fragment across 4 B fragments per
// K-step (4 back-to-back WMMAs with independent accumulators).
// Requires N % 64 == 0, K % 4 == 0, n % 16 == 0.
// ---------------------------------------------------------------------------
template<int K, int N, bool RELU, bool RES>
__global__ __launch_bounds__(256)
void k_gemm_wmma(const float* __restrict__ A, const float* __restrict__ W,
                 const float* __restrict__ bias, const float* __restrict__ res,
                 float* __restrict__ C, int n)
{
  const int CG = N / 64;                   // 16x64 strips per block-row
  int wave = threadIdx.x >> 5;
  int lane = threadIdx.x & 31;
  int tile = blockIdx.x * 8 + wave;
  int total = (n >> 4) * CG;
  if (tile >= total) return;               // wave-uniform guard (EXEC stays full)
  int row0 = (tile / CG) << 4;
  int col0 = (tile % CG) << 6;
  int lh    = lane & 15;
  int hi8   = (lane >> 4) << 3;            // 0 or 8 (C row offset for hi half)
  int khalf = (lane >> 4) << 1;            // 0 or 2 (K offset for hi half)

  const float* arow = A + (size_t)(row0 + lh) * K + khalf;
  const float* w0   = W + (size_t)(col0 + lh) * K + khalf;
  const float* w1   = w0 + (size_t)16 * K;
  const float* w2   = w0 + (size_t)32 * K;
  const float* w3   = w0 + (size_t)48 * K;

  v8f acc0 = {}, acc1 = {}, acc2 = {}, acc3 = {};
  #pragma unroll 4
  for (int k = 0; k < K; k += 4) {
    v2f a, b0, b1, b2, b3;
    a.x  = arow[k]; a.y  = arow[k + 1];
    b0.x = w0[k];   b0.y = w0[k + 1];
    b1.x = w1[k];   b1.y = w1[k + 1];
    b2.x = w2[k];   b2.y = w2[k + 1];
    b3.x = w3[k];   b3.y = w3[k + 1];
    acc0 = __builtin_amdgcn_wmma_f32_16x16x4_f32(false, a, false, b0, (short)0, acc0, false, false);
    acc1 = __builtin_amdgcn_wmma_f32_16x16x4_f32(false, a, false, b1, (short)0, acc1, false, false);
    acc2 = __builtin_amdgcn_wmma_f32_16x16x4_f32(false, a, false, b2, (short)0, acc2, false, false);
    acc3 = __builtin_amdgcn_wmma_f32_16x16x4_f32(false, a, false, b3, (short)0, acc3, false, false);
  }

  int r0 = row0 + hi8;
  v8f accs[4] = {acc0, acc1, acc2, acc3};
  #pragma unroll
  for (int g = 0; g < 4; ++g) {
    int cidx = col0 + g * 16 + lh;
    float bval = bias ? bias[cidx] : 0.0f;
    #pragma unroll
    for (int v = 0; v < 8; ++v) {
      float val = accs[g][v] + bval;
      if (RELU) val = fmaxf(val, 0.0f);
      size_t off2 = (size_t)(r0 + v) * N + cidx;
      if (RES) val += res[off2];
      C[off2] = val;
    }
  }
}

// ---------------------------------------------------------------------------
// CSR build (by destination, includes self loops), deterministic via segment sort
// ---------------------------------------------------------------------------
__global__ void k_zero_ints(int* p, int n) {
  int t = blockIdx.x * blockDim.x + threadIdx.x;
  if (t < n) p[t] = 0;
}

__global__ void k_count_dst(const int* __restrict__ ei, int E, int n, int* __restrict__ cnt) {
  int e = blockIdx.x * blockDim.x + threadIdx.x;
  if (e >= E + n) return;
  int dst = (e < E) ? ei[E + e] : (e - E);
  atomicAdd(&cnt[dst], 1);
}

__global__ __launch_bounds__(1024)
void k_scan_block(const int* __restrict__ cnt, int n, int* __restrict__ excl, int* __restrict__ bsums) {
  __shared__ int sh[1024];
  int gi = blockIdx.x * 1024 + threadIdx.x;
  int v = (gi < n) ? cnt[gi] : 0;
  sh[threadIdx.x] = v; __syncthreads();
  for (int off = 1; off < 1024; off <<= 1) {
    int add = (threadIdx.x >= off) ? sh[threadIdx.x - off] : 0;
    __syncthreads();
    sh[threadIdx.x] += add;
    __syncthreads();
  }
  if (gi < n) excl[gi] = sh[threadIdx.x] - v;     // local exclusive scan
  if (threadIdx.x == 1023) bsums[blockIdx.x] = sh[1023];
}

__global__ void k_scan_sums(int* bsums, int nb) {  // single block of 64 threads
  __shared__ int sh[64];
  int t = threadIdx.x;
  int v = (t < nb) ? bsums[t] : 0;
  sh[t] = v; __syncthreads();
  for (int off = 1; off < 64; off <<= 1) {
    int add = (t >= off) ? sh[t - off] : 0;
    __syncthreads();
    sh[t] += add;
    __syncthreads();
  }
  if (t < nb) bsums[t] = sh[t] - v;                // exclusive
}

__global__ void k_scan_add(const int* __restrict__ excl, const int* __restrict__ bsums,
                           int n, int* __restrict__ rowptr, int total) {
  int gi = blockIdx.x * blockDim.x + threadIdx.x;
  if (gi < n) rowptr[gi] = excl[gi] + bsums[gi >> 10];
  if (gi == 0) rowptr[n] = total;
}

__global__ void k_fill_csr(const int* __restrict__ ei, int E, int n,
                           int* __restrict__ cursor, int* __restrict__ eid) {
  int e = blockIdx.x * blockDim.x + threadIdx.x;
  if (e >= E + n) return;
  int dst = (e < E) ? ei[E + e] : (e - E);
  int pos = atomicAdd(&cursor[dst], 1);
  eid[pos] = e;
}

__global__ void k_sort_segments(const int* __restrict__ rowptr, int* __restrict__ eid,
                                const int* __restrict__ ei, int E,
                                int* __restrict__ csrc, int n) {
  int i = blockIdx.x * blockDim.x + threadIdx.x;
  if (i >= n) return;
  int s = rowptr[i], e = rowptr[i + 1];
  for (int a = s + 1; a < e; ++a) {            // insertion sort by edge id
    int key = eid[a]; int b = a - 1;
    while (b >= s && eid[b] > key) { eid[b + 1] = eid[b]; --b; }
    eid[b + 1] = key;
  }
  for (int a = s; a < e; ++a) {
    int id = eid[a];
    csrc[a] = (id < E) ? ei[id] : (id - E);
  }
}

// ---------------------------------------------------------------------------
// Tiny single-block front: global param norm -> global encoder -> mha row
// ---------------------------------------------------------------------------
__device__ __forceinline__ float block_ln256(float x, float* red, float g, float b) {
  int t = threadIdx.x;
  red[t] = x; __syncthreads();
  for (int off = 128; off > 0; off >>= 1) { if (t < off) red[t] += red[t + off]; __syncthreads(); }
  float mean = red[0] * (1.0f / 256.0f); __syncthreads();
  float d = x - mean; red[t] = d * d; __syncthreads();
  for (int off = 128; off > 0; off >>= 1) { if (t < off) red[t] += red[t + off]; __syncthreads(); }
  float var = red[0] * (1.0f / 256.0f); __syncthreads();
  return d * rsqrtf(var + 1e-5f) * g + b;
}

__global__ __launch_bounds__(256)
void k_front(const float* __restrict__ gp,
             const float* __restrict__ ge_w1, const float* __restrict__ ge_b1,
             const float* __restrict__ ge_g1, const float* __restrict__ ge_be1,
             const float* __restrict__ ge_w2, const float* __restrict__ ge_b2,
             const float* __restrict__ ge_g2, const float* __restrict__ ge_be2,
             const float* __restrict__ in_w, const float* __restrict__ in_b,
             const float* __restrict__ out_w, const float* __restrict__ out_b,
             float* __restrict__ norm_out, float* __restrict__ mha_g)
{
  __shared__ float sn[4], sa[256], sb[256], red[256];
  int t = threadIdx.x;
  if (t < 4) {
    float v = gp[t];
    float nv = (t == 0) ? (v - 4.0f) / 10.0f
             : (t == 1) ? (v - 50.0f) / 250.0f
                        : (v - 60.0f) / 10.0f;
    sn[t] = nv; norm_out[t] = nv;
  }
  __syncthreads();
  float x = ge_b1[t];
  for (int k = 0; k < 4; ++k) x += sn[k] * ge_w1[t * 4 + k];
  x = block_ln256(x, red, ge_g1[t], ge_be1[t]);
  x = fmaxf(x, 0.0f);
  sa[t] = x; __syncthreads();
  float y = ge_b2[t];
  const float* wr = ge_w2 + (size_t)t * 256;
  for (int k = 0; k < 256; ++k) y += sa[k] * wr[k];
  y = block_ln256(y, red, ge_g2[t], ge_be2[t]);
  sb[t] = y; __syncthreads();
  // mha_single_key: out_proj(v_proj(g))
  float v = in_b[512 + t];
  const float* vr = in_w + (size_t)(512 + t) * 256;
  for (int k = 0; k < 256; ++k) v += sb[k] * vr[k];
  sa[t] = v; __syncthreads();
  float o = out_b[t];
  const float* orow = out_w + (size_t)t * 256;
  for (int k = 0; k < 256; ++k) o += sa[k] * orow[k];
  mha_g[t] = o;
}

// ---------------------------------------------------------------------------
// Misc small kernels
// ---------------------------------------------------------------------------
__global__ void k_pad_w7(const float* __restrict__ w, float* __restrict__ wp) { // [256,7]->[256,8]
  int t = blockIdx.x * blockDim.x + threadIdx.x;
  if (t >= 256 * 8) return;
  int r = t >> 3, c = t & 7;
  wp[t] = (c < 7) ? w[r * 7 + c] : 0.0f;
}

__global__ void k_build_xin(const float* __restrict__ x, const float* __restrict__ norm,
                            float* __restrict__ xin, int n) {
  int t = blockIdx.x * blockDim.x + threadIdx.x;
  if (t >= n * 8) return;
  int i = t >> 3, c = t & 7;
  xin[t] = (c < 3) ? x[i * 3 + c] : ((c < 7) ? norm[c - 3] : 0.0f);
}

__global__ void k_att_coef(const float* __restrict__ h, const float* __restrict__ a_s,
                           const float* __restrict__ a_d,
                           float* __restrict__ al_s, float* __restrict__ al_d, int n) {
  int t = blockIdx.x * blockDim.x + threadIdx.x;
  if (t >= n * 4) return;
  int i = t >> 2, hd = t & 3;
  const float* hp = h + (size_t)i * DM + hd * 64;
  const float* as = a_s + hd * 64;
  const float* ad = a_d + hd * 64;
  float ss = 0.0f, sd = 0.0f;
  for (int c = 0; c < 64; ++c) { float hv = hp[c]; ss += hv * as[c]; sd += hv * ad[c]; }
  al_s[t] = ss; al_d[t] = sd;
}

// Per-destination softmax + gather aggregation (no float atomics).
template<bool RELU, bool RES>
__global__ __launch_bounds__(256)
void k_gat_aggregate(const float* __restrict__ h,
                     const float* __restrict__ al_s, const float* __restrict__ al_d,
                     const int* __restrict__ rowptr, const int* __restrict__ csrc,
                     const float* __restrict__ bias, const float* __restrict__ res,
                     float* __restrict__ out)
{
  int i = blockIdx.x;         // destination node
  int t = threadIdx.x;        // channel 0..255
  int hd = t >> 6;
  float ald = al_d[i * 4 + hd];
  int start = rowptr[i], end = rowptr[i + 1];
  float maxe = -1e30f;
  for (int e = start; e < end; ++e) {
    float v = al_s[csrc[e] * 4 + hd] + ald;
    v = (v > 0.0f) ? v : 0.2f * v;
    maxe = fmaxf(maxe, v);
  }
  float den = 0.0f;
  for (int e = start; e < end; ++e) {
    float v = al_s[csrc[e] * 4 + hd] + ald;
    v = (v > 0.0f) ? v : 0.2f * v;
    den += expf(v - maxe);
  }
  float inv = 1.0f / (den + 1e-16f);
  float acc = 0.0f;
  for (int e = start; e < end; ++e) {
    int s = csrc[e];
    float v = al_s[s * 4 + hd] + ald;
    v = (v > 0.0f) ? v : 0.2f * v;
    acc += expf(v - maxe) * inv * h[(size_t)s * DM + t];
  }
  float val = acc + bias[t];
  if (RELU) val = fmaxf(val, 0.0f);
  size_t off = (size_t)i * DM + t;
  if (RES) val += res[off];
  out[off] = val;
}

__global__ __launch_bounds__(256)
void k_colsum_partial(const float* __restrict__ x, int n, float* __restrict__ part) {
  int t = threadIdx.x;
  int chunk = (n + gridDim.x - 1) / gridDim.x;
  int i0 = blockIdx.x * chunk;
  int i1 = i0 + chunk; if (i1 > n) i1 = n;
  float s = 0.0f;
  for (int i = i0; i < i1; ++i) s += x[(size_t)i * DM + t];
  part[blockIdx.x * DM + t] = s;
}

__global__ __launch_bounds__(256)
void k_mean_mha(const float* __restrict__ part, int PB, int n,
                const float* __restrict__ in_w, const float* __restrict__ in_b,
                const float* __restrict__ out_w, const float* __restrict__ out_b,
                float* __restrict__ mha_mean)
{
  __shared__ float sm[256], sv[256];
  int t = threadIdx.x;
  float s = 0.0f;
  for (int b = 0; b < PB; ++b) s += part[b * DM + t];
  sm[t] = s / (float)n; __syncthreads();
  float v = in_b[512 + t];
  const float* wr = in_w + (size_t)(512 + t) * 256;
  for (int k = 0; k < 256; ++k) v += sm[k] * wr[k];
  sv[t] = v; __syncthreads();
  float o = out_b[t];
  const float* orow = out_w + (size_t)t * 256;
  for (int k = 0; k < 256; ++k) o += sv[k] * orow[k];
  mha_mean[t] = o;
}

__global__ __launch_bounds__(256)
void k_ln_rows_add(const float* __restrict__ x, const float* __restrict__ addrow,
                   const float* __restrict__ g, const float* __restrict__ b,
                   float* __restrict__ out, int n)
{
  __shared__ float red[256];
  int i = blockIdx.x, t = threadIdx.x;
  float v = x[(size_t)i * DM + t] + addrow[t];
  red[t] = v; __syncthreads();
  for (int off = 128; off > 0; off >>= 1) { if (t < off) red[t] += red[t + off]; __syncthreads(); }
  float mean = red[0] * (1.0f / DM); __syncthreads();
  float d = v - mean; red[t] = d * d; __syncthreads();
  for (int off = 128; off > 0; off >>= 1) { if (t < off) red[t] += red[t + off]; __syncthreads(); }
  float var = red[0] * (1.0f / DM);
  out[(size_t)i * DM + t] = d * rsqrtf(var + 1e-5f) * g[t] + b[t];
}

__global__ __launch_bounds__(64)
void k_ln64_relu(const float* __restrict__ x, const float* __restrict__ g,
                 const float* __restrict__ b, float* __restrict__ out, int n)
{
  __shared__ float red[64];
  int i = blockIdx.x, t = threadIdx.x;
  float v = x[(size_t)i * 64 + t];
  red[t] = v; __syncthreads();
  for (int off = 32; off > 0; off >>= 1) { if (t < off) red[t] += red[t + off]; __syncthreads(); }
  float mean = red[0] * (1.0f / 64.0f); __syncthreads();
  float d = v - mean; red[t] = d * d; __syncthreads();
  for (int off = 32; off > 0; off >>= 1) { if (t < off) red[t] += red[t + off]; __syncthreads(); }
  float var = red[0] * (1.0f / 64.0f);
  float r = d * rsqrtf(var + 1e-5f) * g[t] + b[t];
  out[(size_t)i * 64 + t] = fmaxf(r, 0.0f);
}

__global__ void k_lin3(const float* __restrict__ h, const float* __restrict__ w,
                       const float* __restrict__ b, float* __restrict__ out, int n) {
  int t = blockIdx.x * blockDim.x + threadIdx.x;
  if (t >= n * 3) return;
  int i = t / 3, o = t - i * 3;
  float s = b[o];
  const float* wr = w + o * 64;
  const float* hr = h + (size_t)i * 64;
  for (int k = 0; k < 64; ++k) s += hr[k] * wr[k];
  out[t] = s;
}

__global__ void k_add3(const float* __restrict__ x, const float* __restrict__ f,
                       float* __restrict__ d, int n3) {
  int t = blockIdx.x * blockDim.x + threadIdx.x;
  if (t < n3) d[t] = x[t] + f[t];
}

// ---------------------------------------------------------------------------
extern "C" void kernel_launch(void* const* d_in, const int* in_sizes, int n_in,
                              void* d_out, int out_size, void* d_ws, size_t ws_size,
                              hipStream_t stream)
{
  const float* x   = (const float*)d_in[0];
  const int*   ei  = (const int*)d_in[1];
  const float* gp  = (const float*)d_in[2];
  const int n = in_sizes[0] / 3;
  const int E = in_sizes[1] / 2;
  const int ET = E + n;

  const float* W1    = (const float*)d_in[3];
  const float* as1   = (const float*)d_in[4];
  const float* ad1   = (const float*)d_in[5];
  const float* b1    = (const float*)d_in[6];
  const float* W2    = (const float*)d_in[7];
  const float* as2   = (const float*)d_in[8];
  const float* ad2   = (const float*)d_in[9];
  const float* b2    = (const float*)d_in[10];
  const float* W3    = (const float*)d_in[11];
  const float* as3   = (const float*)d_in[12];
  const float* ad3   = (const float*)d_in[13];
  const float* b3    = (const float*)d_in[14];
  const float* ge_w1 = (const float*)d_in[15];
  const float* ge_b1 = (const float*)d_in[16];
  const float* ge_g1 = (const float*)d_in[17];
  const float* ge_be1= (const float*)d_in[18];
  const float* ge_w2 = (const float*)d_in[19];
  const float* ge_b2 = (const float*)d_in[20];
  const float* ge_g2 = (const float*)d_in[21];
  const float* ge_be2= (const float*)d_in[22];
  const float* skip_w= (const float*)d_in[23];
  const float* skip_b= (const float*)d_in[24];
  const float* in_w  = (const float*)d_in[25];
  const float* in_b  = (const float*)d_in[26];
  const float* out_w = (const float*)d_in[27];
  const float* out_b = (const float*)d_in[28];
  const float* fn_g  = (const float*)d_in[29];
  const float* fn_b  = (const float*)d_in[30];
  const float* fp_w  = (const float*)d_in[31];
  const float* fp_b  = (const float*)d_in[32];
  const float* fd_w1 = (const float*)d_in[33];
  const float* fd_b1 = (const float*)d_in[34];
  const float* fd_g1 = (const float*)d_in[35];
  const float* fd_be1= (const float*)d_in[36];
  const float* fd_w2 = (const float*)d_in[37];
  const float* fd_b2 = (const float*)d_in[38];
  const float* fd_g2 = (const float*)d_in[39];
  const float* fd_be2= (const float*)d_in[40];
  const float* fd_w3 = (const float*)d_in[41];
  const float* fd_b3 = (const float*)d_in[42];
  const float* rd_w1 = (const float*)d_in[43];
  const float* rd_b1 = (const float*)d_in[44];
  const float* rd_g1 = (const float*)d_in[45];
  const float* rd_be1= (const float*)d_in[46];
  const float* rd_w2 = (const float*)d_in[47];
  const float* rd_b2 = (const float*)d_in[48];
  const float* rd_g2 = (const float*)d_in[49];
  const float* rd_be2= (const float*)d_in[50];
  const float* rd_w3 = (const float*)d_in[51];
  const float* rd_b3 = (const float*)d_in[52];

  // ---- workspace carve ----
  char* wsb = (char*)d_ws;
  size_t off = 0;
  auto carve = [&](size_t bytes) -> void* {
    void* p = wsb + off;
    off += (bytes + 255) & ~(size_t)255;
    return p;
  };
  float* f_norm = (float*)carve(4 * 4);
  float* f_mhag = (float*)carve(256 * 4);
  float* f_mham = (float*)carve(256 * 4);
  float* f_part = (float*)carve(128 * 256 * 4);
  float* f_w1p  = (float*)carve(256 * 8 * 4);
  float* f_skp  = (float*)carve(256 * 8 * 4);
  float* f_xin  = (float*)carve((size_t)n * 8 * 4);
  float* f_als  = (float*)carve((size_t)n * 4 * 4);
  float* f_ald  = (float*)carve((size_t)n * 4 * 4);
  float* f_d1   = (float*)carve((size_t)n * 64 * 4);
  float* f_d2   = (float*)carve((size_t)n * 64 * 4);
  float* f_def  = (float*)carve((size_t)n * 3 * 4);
  int*   i_cnt  = (int*)carve((size_t)n * 4);
  int*   i_excl = (int*)carve((size_t)n * 4);
  int*   i_rowp = (int*)carve((size_t)(n + 1) * 4);
  int*   i_cur  = (int*)carve((size_t)(n + 1) * 4);
  int*   i_bs   = (int*)carve(64 * 4);
  int*   i_eid  = (int*)carve((size_t)ET * 4);
  int*   i_csrc = (int*)carve((size_t)ET * 4);
  float* bufS   = (float*)carve((size_t)n * DM * 4);
  float* bufH   = (float*)carve((size_t)n * DM * 4);
  float* buf1   = (float*)carve((size_t)n * DM * 4);
  float* buf2   = (float*)carve((size_t)n * DM * 4);

  // ---- CSR build (graph is static; rebuilt deterministically each call) ----
  k_zero_ints<<<(n + 255) / 256, 256, 0, stream>>>(i_cnt, n);
  k_count_dst<<<(ET + 255) / 256, 256, 0, stream>>>(ei, E, n, i_cnt);
  int nsb = (n + 1023) / 1024;
  k_scan_block<<<nsb, 1024, 0, stream>>>(i_cnt, n, i_excl, i_bs);
  k_scan_sums<<<1, 64, 0, stream>>>(i_bs, nsb);
  k_scan_add<<<(n + 255) / 256, 256, 0, stream>>>(i_excl, i_bs, n, i_rowp, ET);
  hipMemcpyAsync(i_cur, i_rowp, (size_t)n * 4, hipMemcpyDeviceToDevice, stream);
  k_fill_csr<<<(ET + 255) / 256, 256, 0, stream>>>(ei, E, n, i_cur, i_eid);
  k_sort_segments<<<(n + 255) / 256, 256, 0, stream>>>(i_rowp, i_eid, ei, E, i_csrc, n);

  // ---- global-param front path (done once, reused by both encodes) ----
  k_front<<<1, 256, 0, stream>>>(gp, ge_w1, ge_b1, ge_g1, ge_be1,
                                 ge_w2, ge_b2, ge_g2, ge_be2,
                                 in_w, in_b, out_w, out_b, f_norm, f_mhag);
  k_pad_w7<<<(2048 + 255) / 256, 256, 0, stream>>>(W1, f_w1p);
  k_pad_w7<<<(2048 + 255) / 256, 256, 0, stream>>>(skip_w, f_skp);

  const int gb256 = ((n / 16) * 4 + 7) / 8;   // N=256: 4 strips of 16x64 per row
  const int gb64  = ((n / 16) * 1 + 7) / 8;   // N=64: 1 strip per row

  auto encode = [&](const float* coords) -> float* {
    k_build_xin<<<(n * 8 + 255) / 256, 256, 0, stream>>>(coords, f_norm, f_xin, n);
    // skip projection
    k_gemm_wmma<8, 256, false, false><<<gb256, 256, 0, stream>>>(f_xin, f_skp, skip_b, nullptr, bufS, n);
    // GAT1: relu(gat) + x_skip
    k_gemm_wmma<8, 256, false, false><<<gb256, 256, 0, stream>>>(f_xin, f_w1p, nullptr, nullptr, bufH, n);
    k_att_coef<<<(n * 4 + 255) / 256, 256, 0, stream>>>(bufH, as1, ad1, f_als, f_ald, n);
    k_gat_aggregate<true, true><<<n, 256, 0, stream>>>(bufH, f_als, f_ald, i_rowp, i_csrc, b1, bufS, buf1);
    // GAT2: relu(gat) + x1
    k_gemm_wmma<256, 256, false, false><<<gb256, 256, 0, stream>>>(buf1, W2, nullptr, nullptr, bufH, n);
    k_att_coef<<<(n * 4 + 255) / 256, 256, 0, stream>>>(bufH, as2, ad2, f_als, f_ald, n);
    k_gat_aggregate<true, true><<<n, 256, 0, stream>>>(bufH, f_als, f_ald, i_rowp, i_csrc, b2, buf1, buf2);
    // GAT3: gat + x2
    k_gemm_wmma<256, 256, false, false><<<gb256, 256, 0, stream>>>(buf2, W3, nullptr, nullptr, bufH, n);
    k_att_coef<<<(n * 4 + 255) / 256, 256, 0, stream>>>(bufH, as3, ad3, f_als, f_ald, n);
    k_gat_aggregate<false, true><<<n, 256, 0, stream>>>(bufH, f_als, f_ald, i_rowp, i_csrc, b3, buf2, bufS);
    // mean pool + mha(mean row)
    k_colsum_partial<<<128, 256, 0, stream>>>(bufS, n, f_part);
    k_mean_mha<<<1, 256, 0, stream>>>(f_part, 128, n, in_w, in_b, out_w, out_b, f_mham);
    // fusion layernorms + final projection
    k_ln_rows_add<<<n, 256, 0, stream>>>(bufS, f_mhag, fn_g, fn_b, bufH, n);
    k_ln_rows_add<<<n, 256, 0, stream>>>(bufH, f_mham, fn_g, fn_b, buf1, n);
    k_gemm_wmma<256, 256, false, false><<<gb256, 256, 0, stream>>>(buf1, fp_w, fp_b, nullptr, buf2, n);
    return buf2;
  };

  auto decode = [&](const float* z,
                    const float* w1, const float* db1, const float* g1, const float* be1,
                    const float* w2, const float* db2, const float* g2, const float* be2,
                    const float* w3, const float* db3, float* out3) {
    k_gemm_wmma<256, 64, false, false><<<gb64, 256, 0, stream>>>(z, w1, db1, nullptr, f_d1, n);
    k_ln64_relu<<<n, 64, 0, stream>>>(f_d1, g1, be1, f_d2, n);
    k_gemm_wmma<64, 64, false, false><<<gb64, 256, 0, stream>>>(f_d2, w2, db2, nullptr, f_d1, n);
    k_ln64_relu<<<n, 64, 0, stream>>>(f_d1, g2, be2, f_d2, n);
    k_lin3<<<(n * 3 + 255) / 256, 256, 0, stream>>>(f_d2, w3, db3, out3, n);
  };

  float* fwd = (float*)d_out;
  float* rev = (float*)d_out + (size_t)n * 3;

  float* enc1 = encode(x);
  decode(enc1, fd_w1, fd_b1, fd_g1, fd_be1, fd_w2, fd_b2, fd_g2, fd_be2, fd_w3, fd_b3, fwd);
  k_add3<<<(n * 3 + 255) / 256, 256, 0, stream>>>(x, fwd, f_def, n * 3);
  float* enc2 = encode(f_def);
  decode(enc2, rd_w1, rd_b1, rd_g1, rd_be1, rd_w2, rd_b2, rd_g2, rd_be2, rd_w3, rd_b3, rev);
}